// ProbAttention_29119878266974
// MI455X (gfx1250) — compile-verified
//
#include <hip/hip_runtime.h>
#include <hip/hip_bf16.h>
#include <math.h>
#include <stdint.h>

// Problem constants (fixed by the reference setup)
constexpr int B_ = 4;
constexpr int L_ = 2048;
constexpr int H_ = 8;
constexpr int D_ = 64;
constexpr int S_ = 40;   // sample_k
constexpr int U_ = 40;   // top-k count
constexpr int UP = 48;   // U padded to 3 WMMA M-tiles
constexpr int CH = 32;   // key chunk size in the attention kernel

typedef __attribute__((ext_vector_type(2))) float v2f;
typedef __attribute__((ext_vector_type(8))) float v8f;

static __device__ __forceinline__ v8f wmma_f32_4(v2f a, v2f b, v8f c) {
    // V_WMMA_F32_16X16X4_F32 : D = A(16x4 f32) * B(4x16 f32) + C(16x16 f32)
    return __builtin_amdgcn_wmma_f32_16x16x4_f32(
        /*neg_a=*/false, a, /*neg_b=*/false, b,
        /*c_mod=*/(short)0, c, /*reuse_a=*/false, /*reuse_b=*/false);
}

// ---------------------------------------------------------------------------
// Async global->LDS 16B copy (GLOBAL_LOAD_ASYNC_TO_LDS_B128, tracked by
// ASYNCcnt) with guarded fallback to a plain float4 round trip.
// Builtin signature (from hipcc diagnostic): (V4i addrspace(1)*, V4i
// addrspace(3)*, imm int offset, imm int cpol).
// ---------------------------------------------------------------------------
#if __has_builtin(__builtin_amdgcn_global_load_async_to_lds_b128) && \
    __has_builtin(__builtin_amdgcn_s_wait_asynccnt)
#define ASYNC_LDS 1
#endif

typedef int v4i __attribute__((vector_size(4 * sizeof(int))));
typedef __attribute__((address_space(1))) v4i* as1_v4i_p;
typedef __attribute__((address_space(3))) v4i* as3_v4i_p;

static __device__ __forceinline__ void cp16(const float* __restrict__ g, float* l) {
#ifdef ASYNC_LDS
    __builtin_amdgcn_global_load_async_to_lds_b128(
        (as1_v4i_p)(v4i*)(uintptr_t)g,   // strip const, retype, cast to AS1
        (as3_v4i_p)(v4i*)l,              // generic -> AS3 (LDS)
        0, 0);
#else
    *(float4*)l = *(const float4*)g;
#endif
}
static __device__ __forceinline__ void cp16_wait() {
#ifdef ASYNC_LDS
    __builtin_amdgcn_s_wait_asynccnt(0);
#endif
}

// ---------------------------------------------------------------------------
// Kernel 1: M[b,h,l] = max_s(Q_l . K_idx[l,s]) - (1/L) * sum_s(Q_l . K_idx[l,s])
// One wave (32 lanes) per (b,h,l); lane owns 2 of the 64 D elements.
// ---------------------------------------------------------------------------
__global__ __launch_bounds__(256) void kernM(const float* __restrict__ Q,
                                             const float* __restrict__ K,
                                             const int* __restrict__ idx,
                                             float* __restrict__ Mws) {
    int gw   = (blockIdx.x * blockDim.x + threadIdx.x) >> 5;  // global wave id
    int lane = threadIdx.x & 31;
    int l  = gw % L_;
    int bh = gw / L_;
    int b  = bh >> 3;
    int h  = bh & 7;

    const float* qp = Q + (((size_t)b * L_ + l) * H_ + h) * D_;
    float2 q = *(const float2*)(qp + lane * 2);

    float mx = -3.0e38f;
    float sm = 0.0f;
    for (int s = 0; s < S_; ++s) {
        int kr = idx[l * S_ + s];
        const float* kp = K + (((size_t)b * L_ + kr) * H_ + h) * D_;
        float2 kv = *(const float2*)(kp + lane * 2);
        float p = q.x * kv.x + q.y * kv.y;
        // full-wave butterfly sum -> every lane holds the dot product
        for (int m = 16; m > 0; m >>= 1) p += __shfl_xor(p, m, 32);
        mx = fmaxf(mx, p);
        sm += p;
    }
    if (lane == 0) Mws[(size_t)bh * L_ + l] = mx - sm * (1.0f / (float)L_);
}

// ---------------------------------------------------------------------------
// Kernel 2: per (b,h) top-40 of M (strict-> keeps lowest index on ties, like
// lax.top_k), store indices, gather Q_reduce (padded to 48 rows with zeros).
// ---------------------------------------------------------------------------
__global__ __launch_bounds__(256) void kernTopK(const float* __restrict__ Mws,
                                                const float* __restrict__ Q,
                                                int* __restrict__ Mtopws,
                                                float* __restrict__ Qrws) {
    int bh = blockIdx.x;
    int b  = bh >> 3;
    int h  = bh & 7;
    int tid = threadIdx.x;

    __shared__ float vals[L_];
    __shared__ float rv[256];
    __shared__ int   ri[256];
    __shared__ int   topIdx[UP];

    for (int i = tid; i < L_; i += 256) vals[i] = Mws[(size_t)bh * L_ + i];
    if (tid < UP) topIdx[tid] = 0;
    __syncthreads();

    for (int round = 0; round < U_; ++round) {
        float best = -3.4e38f;
        int   bi   = 0;
        for (int i = tid; i < L_; i += 256) {
            float v = vals[i];
            if (v > best) { best = v; bi = i; }
        }
        rv[tid] = best;
        ri[tid] = bi;
        __syncthreads();
        for (int s = 128; s > 0; s >>= 1) {
            if (tid < s) {
                if (rv[tid + s] > rv[tid] ||
                    (rv[tid + s] == rv[tid] && ri[tid + s] < ri[tid])) {
                    rv[tid] = rv[tid + s];
                    ri[tid] = ri[tid + s];
                }
            }
            __syncthreads();
        }
        if (tid == 0) {
            topIdx[round] = ri[0];
            vals[ri[0]]   = -3.4e38f;
        }
        __syncthreads();
    }

    if (tid < UP) Mtopws[bh * UP + tid] = topIdx[tid];

    // Gather Q_reduce (48 x 64); rows >= 40 zero-padded.
    for (int i = tid; i < UP * D_; i += 256) {
        int m = i >> 6;
        int d = i & 63;
        float v = 0.0f;
        if (m < U_) v = Q[(((size_t)b * L_ + topIdx[m]) * H_ + h) * D_ + d];
        Qrws[(size_t)bh * UP * D_ + i] = v;
    }
}

// ---------------------------------------------------------------------------
// Kernel 3a: V_mean[bh][d] = mean over L of V[b,:,h,d]   (-> workspace)
// ---------------------------------------------------------------------------
__global__ __launch_bounds__(256) void kernMean(const float* __restrict__ V,
                                                float* __restrict__ Vmws) {
    int bh = blockIdx.x;
    int b  = bh >> 3;
    int h  = bh & 7;
    int tid = threadIdx.x;

    __shared__ float part[4][D_];
    int d  = tid & 63;
    int sl = tid >> 6;  // 4 slices of 512 rows
    float s = 0.0f;
    for (int l = sl * (L_ / 4); l < (sl + 1) * (L_ / 4); ++l)
        s += V[(((size_t)b * L_ + l) * H_ + h) * D_ + d];
    part[sl][d] = s;
    __syncthreads();
    if (tid < D_)
        Vmws[bh * D_ + tid] =
            (part[0][tid] + part[1][tid] + part[2][tid] + part[3][tid]) *
            (1.0f / (float)L_);
}

// ---------------------------------------------------------------------------
// Kernel 3b: broadcast fill of the output with V_mean. 512 blocks so the
// 16 MB write stream can actually occupy the machine.
// ---------------------------------------------------------------------------
__global__ __launch_bounds__(256) void kernFill(const float* __restrict__ Vmws,
                                                float* __restrict__ out) {
    int blk = blockIdx.x;
    int bh  = blk >> 4;
    int seg = blk & 15;  // 16 segments of 128 rows
    int tid = threadIdx.x;

    __shared__ float vm[D_];
    if (tid < D_) vm[tid] = Vmws[bh * D_ + tid];
    __syncthreads();

    size_t base = (size_t)bh * L_ * D_ + (size_t)seg * 128 * D_;
    for (int i = tid; i < 128 * (D_ / 4); i += 256) {
        int row = i >> 4;
        int c4  = (i & 15) * 4;
        *(float4*)&out[base + row * D_ + c4] = *(float4*)&vm[c4];
    }
}

// ---------------------------------------------------------------------------
// Kernel 4: dense attention for the 40 selected queries, per (b,h).
// Two-pass softmax over 32-key chunks; both GEMMs via V_WMMA_F32_16X16X4_F32.
// 4 waves: waves 0..2 own the three 16-row M-tiles of scores (two N-subtiles
// per chunk); all 4 waves own one 16-wide D column each in P@V.
// ---------------------------------------------------------------------------
__global__ __launch_bounds__(128) void kernAttn(const float* __restrict__ K,
                                                const float* __restrict__ V,
                                                const float* __restrict__ Qr,
                                                const int* __restrict__ Mtop,
                                                float* __restrict__ out) {
    constexpr int QS = 68;  // LDS row stride (floats): 68 % 64 = 4 -> conflict-free
    constexpr int PS = 36;  // 32 + 4 pad: gcd(36,64)=4 -> conflict-free A reads

    __shared__ float sQ[UP][QS];
    __shared__ float sK[CH][QS];
    __shared__ float sV[CH][QS];
    __shared__ float sP[UP][PS];
    __shared__ float rowmax[UP];
    __shared__ float rowsum[UP];

    int bh = blockIdx.x;
    int b  = bh >> 3;
    int h  = bh & 7;
    int tid  = threadIdx.x;
    int wave = tid >> 5;
    int lane = tid & 31;
    int half = lane >> 4;   // which 16-lane half
    int r    = lane & 15;

    const float scale = 0.125f;  // 1/sqrt(64)

    // Load Q_reduce tile (48 x 64) once (async -> LDS).
    for (int i = tid; i < UP * (D_ / 4); i += 128) {
        int row = i >> 4;
        int c4  = (i & 15) * 4;
        cp16(&Qr[(size_t)bh * UP * D_ + row * D_ + c4], &sQ[row][c4]);
    }
    if (tid < UP) { rowmax[tid] = -3.0e38f; rowsum[tid] = 0.0f; }
    cp16_wait();
    __syncthreads();

    // ------------------ PASS 1: per-row max of scaled scores ------------------
    for (int ch = 0; ch < L_ / CH; ++ch) {
        for (int i = tid; i < CH * (D_ / 4); i += 128) {
            int row = i >> 4;
            int c4  = (i & 15) * 4;
            cp16(&K[(((size_t)b * L_ + ch * CH + row) * H_ + h) * D_ + c4],
                 &sK[row][c4]);
        }
        if (ch + 1 < L_ / CH && tid < CH)  // prefetch next chunk row starts
            __builtin_prefetch(&K[(((size_t)b * L_ + (ch + 1) * CH + tid) * H_ + h) * D_], 0, 0);
        cp16_wait();
        __syncthreads();

        if (wave < 3) {
            for (int nt = 0; nt < 2; ++nt) {
                v8f acc = {0.f, 0.f, 0.f, 0.f, 0.f, 0.f, 0.f, 0.f};
                for (int kk = 0; kk < 16; ++kk) {
                    int c = kk * 4 + half * 2;
                    v2f a, bb;
                    a.x  = sQ[wave * 16 + r][c];
                    a.y  = sQ[wave * 16 + r][c + 1];
                    bb.x = sK[nt * 16 + r][c];  // B[k][n]: n = key = r, k = D comp
                    bb.y = sK[nt * 16 + r][c + 1];
                    acc = wmma_f32_4(a, bb, acc);
                }
                for (int j = 0; j < 8; ++j) {
                    float v = acc[j] * scale;
                    for (int m = 8; m > 0; m >>= 1) v = fmaxf(v, __shfl_xor(v, m, 32));
                    if (r == 0) {
                        int row = wave * 16 + j + half * 8;
                        rowmax[row] = fmaxf(rowmax[row], v);
                    }
                }
            }
        }
        __syncthreads();
    }

    // ------------------ PASS 2: exp / row-sum / P @ V -------------------------
    v8f ctx0 = {0.f, 0.f, 0.f, 0.f, 0.f, 0.f, 0.f, 0.f};
    v8f ctx1 = {0.f, 0.f, 0.f, 0.f, 0.f, 0.f, 0.f, 0.f};
    v8f ctx2 = {0.f, 0.f, 0.f, 0.f, 0.f, 0.f, 0.f, 0.f};

    for (int ch = 0; ch < L_ / CH; ++ch) {
        for (int i = tid; i < CH * (D_ / 4); i += 128) {
            int row = i >> 4;
            int c4  = (i & 15) * 4;
            size_t gbase = (((size_t)b * L_ + ch * CH + row) * H_ + h) * D_ + c4;
            cp16(&K[gbase], &sK[row][c4]);
            cp16(&V[gbase], &sV[row][c4]);
        }
        if (ch + 1 < L_ / CH && tid < CH) {
            __builtin_prefetch(&K[(((size_t)b * L_ + (ch + 1) * CH + tid) * H_ + h) * D_], 0, 0);
            __builtin_prefetch(&V[(((size_t)b * L_ + (ch + 1) * CH + tid) * H_ + h) * D_], 0, 0);
        }
        cp16_wait();
        __syncthreads();

        if (wave < 3) {
            for (int nt = 0; nt < 2; ++nt) {
                v8f acc = {0.f, 0.f, 0.f, 0.f, 0.f, 0.f, 0.f, 0.f};
                for (int kk = 0; kk < 16; ++kk) {
                    int c = kk * 4 + half * 2;
                    v2f a, bb;
                    a.x  = sQ[wave * 16 + r][c];
                    a.y  = sQ[wave * 16 + r][c + 1];
                    bb.x = sK[nt * 16 + r][c];
                    bb.y = sK[nt * 16 + r][c + 1];
                    acc = wmma_f32_4(a, bb, acc);
                }
                for (int j = 0; j < 8; ++j) {
                    int row = wave * 16 + j + half * 8;
                    float p = expf(acc[j] * scale - rowmax[row]);
                    sP[row][nt * 16 + r] = p;  // A-layout staging for GEMM2
                    float s = p;
                    for (int m = 8; m > 0; m >>= 1) s += __shfl_xor(s, m, 32);
                    if (r == 0) rowsum[row] += s;
                }
            }
        }
        __syncthreads();

        // GEMM2: ctx += P(48x32) @ V(32x64); this wave owns D cols
        // [wave*16, wave*16+16). 8 K-steps cover the 32 keys of the chunk.
        {
            int n = wave * 16 + r;  // D column handled by lane (within wave tile)
            for (int kk = 0; kk < 8; ++kk) {
                int c = kk * 4 + half * 2;  // key index within chunk
                v2f bb;
                bb.x = sV[c][n];
                bb.y = sV[c + 1][n];
                v2f a0, a1, a2;
                a0.x = sP[0 * 16 + r][c];  a0.y = sP[0 * 16 + r][c + 1];
                a1.x = sP[1 * 16 + r][c];  a1.y = sP[1 * 16 + r][c + 1];
                a2.x = sP[2 * 16 + r][c];  a2.y = sP[2 * 16 + r][c + 1];
                ctx0 = wmma_f32_4(a0, bb, ctx0);
                ctx1 = wmma_f32_4(a1, bb, ctx1);
                ctx2 = wmma_f32_4(a2, bb, ctx2);
            }
        }
        __syncthreads();
    }

    // ------------------ Finalize: divide by row sums, scatter to M_top --------
    for (int mt = 0; mt < 3; ++mt) {
        for (int j = 0; j < 8; ++j) {
            int row = mt * 16 + j + half * 8;
            if (row < U_) {
                float val = (mt == 0 ? ctx0[j] : (mt == 1 ? ctx1[j] : ctx2[j]));
                val /= rowsum[row];
                int lq = Mtop[bh * UP + row];
                out[((size_t)bh * L_ + lq) * D_ + wave * 16 + r] = val;
            }
        }
    }
}

// ---------------------------------------------------------------------------
extern "C" void kernel_launch(void* const* d_in, const int* in_sizes, int n_in,
                              void* d_out, int out_size, void* d_ws, size_t ws_size,
                              hipStream_t stream) {
    (void)in_sizes; (void)n_in; (void)out_size; (void)ws_size;

    const float* Q   = (const float*)d_in[0];  // (B,L,H,D) f32
    const float* K   = (const float*)d_in[1];  // (B,L,H,D) f32
    const float* V   = (const float*)d_in[2];  // (B,L,H,D) f32
    const int*   idx = (const int*)d_in[3];    // (L,S) i32
    float* out = (float*)d_out;                // (B,H,L,D) f32

    // Workspace layout (< 700 KB total, all 16B-aligned offsets)
    float* Mws    = (float*)d_ws;                        // B*H*L floats
    int*   Mtopws = (int*)(Mws + (size_t)B_ * H_ * L_);  // B*H*48 ints
    float* Qrws   = (float*)(Mtopws + B_ * H_ * UP);     // B*H*48*64 floats
    float* Vmws   = Qrws + (size_t)B_ * H_ * UP * D_;    // B*H*64 floats

    // 1) sparsity measurement M
    kernM<<<(B_ * H_ * L_) / 8, 256, 0, stream>>>(Q, K, idx, Mws);
    // 2) top-40 + Q_reduce gather
    kernTopK<<<B_ * H_, 256, 0, stream>>>(Mws, Q, Mtopws, Qrws);
    // 3) V-mean + wide broadcast fill (attention rows overwritten next)
    kernMean<<<B_ * H_, 256, 0, stream>>>(V, Vmws);
    kernFill<<<B_ * H_ * 16, 256, 0, stream>>>(Vmws, out);
    // 4) WMMA attention for the selected queries + scatter
    kernAttn<<<B_ * H_, 128, 0, stream>>>(K, V, Qrws, Mtopws, out);
}